// FIND_LOCATION_43980465111763
// MI455X (gfx1250) — compile-verified
//
#include <hip/hip_runtime.h>
#include <hip/hip_bf16.h>

typedef __attribute__((ext_vector_type(2))) float v2f;  // A/B for 16x16x4 f32: 64 f32 / 32 lanes
typedef __attribute__((ext_vector_type(8))) float v8f;  // C/D 16x16 f32 accumulator

// Sum x across all 32 lanes of the wave using V_WMMA_F32_16X16X4_F32.
// A holds each lane's value in one of the 64 A-matrix slots (bijective VGPR
// mapping), B is all-ones so D[m][n] = rowsum_m(A) regardless of A's exact
// (lane,vgpr)->(M,K) mapping. Per documented D layout, lanes 0-15 hold rows
// 0-7 and lanes 16-31 hold rows 8-15; sum 8 D VGPRs + one shfl_xor(16)
// => total in every lane. EXEC must be all-ones at call sites (it is: the
// calls are either unconditional or under a wave-uniform branch).
__device__ __forceinline__ float wave_sum_wmma(float x) {
  v2f a; a.x = x;    a.y = 0.0f;
  v2f b; b.x = 1.0f; b.y = 1.0f;
  v8f c = {};
  // 8 args: (neg_a, A, neg_b, B, c_mod, C, reuse_a, reuse_b); neg must be 0 for f32
  v8f d = __builtin_amdgcn_wmma_f32_16x16x4_f32(
      false, a, false, b, (short)0, c, false, false);
  float s = d[0] + d[1] + d[2] + d[3] + d[4] + d[5] + d[6] + d[7];
  s += __shfl_xor(s, 16, 32);  // combine row-block 0-7 with row-block 8-15
  return s;
}

__global__ __launch_bounds__(1024) void FIND_LOCATION_43980465111763_kernel(
    const float* __restrict__ lat_p, const float* __restrict__ lon_p,
    const float* __restrict__ v_p,   const float* __restrict__ st_lat,
    const float* __restrict__ st_lon, const float* __restrict__ times,
    float* __restrict__ out, int N) {
  const float DEGf = (float)(3.14 / 180.0);  // module uses 3.14, not pi
  const float lat = lat_p[0];
  const float lon = lon_p[0];
  const float v   = v_p[0];

  const float la2 = lat * DEGf;
  const float lo2 = lon * DEGf;
  const float cos_la2 = cosf(la2);

  // Per-lane partial sums of r and r^2, r_i = d_i - v*t_i
  float sR = 0.0f, sQ = 0.0f;
  for (int i = threadIdx.x; i < N; i += blockDim.x) {
    float la1  = st_lat[i] * DEGf;
    float lo1  = st_lon[i] * DEGf;
    float sdla = sinf((la2 - la1) * 0.5f);
    float sdlo = sinf((lo2 - lo1) * 0.5f);
    float a    = sdla * sdla + cosf(la1) * cos_la2 * sdlo * sdlo;
    float cc   = 2.0f * atan2f(sqrtf(a), sqrtf(1.0f - a));
    float dist = 6373.0f * cc;
    float r    = dist - v * times[i];
    sR += r;
    sQ += r * r;
  }

  // Intra-wave reduction via WMMA (all waves execute; EXEC all ones)
  float wR = wave_sum_wmma(sR);
  float wQ = wave_sum_wmma(sQ);

  __shared__ float shR[32];
  __shared__ float shQ[32];
  const int lane   = threadIdx.x & 31;
  const int wave   = threadIdx.x >> 5;
  const int nwaves = (blockDim.x + 31) >> 5;
  if (lane == 0) { shR[wave] = wR; shQ[wave] = wQ; }
  __syncthreads();

  if (wave == 0) {  // wave-uniform branch: EXEC all ones inside
    float xR = (lane < nwaves) ? shR[lane] : 0.0f;
    float xQ = (lane < nwaves) ? shQ[lane] : 0.0f;
    float tR = wave_sum_wmma(xR);
    float tQ = wave_sum_wmma(xQ);
    if (lane == 0) {
      // pair_sum over i<j of (r_j - r_i)^2 == N*Sum(r^2) - (Sum r)^2
      double SR = (double)tR;
      double SQ = (double)tQ;
      double pair_sum = (double)N * SQ - SR * SR;
      double num = (double)N * (double)(N - 1) * 0.5;
      double loss = ((v < 0.0f) ? (double)(-v) * 10.0 : 0.0) + pair_sum;
      loss /= num;
      double dv = (double)v - 6.0;
      if (fabs(dv) > 4.0) loss += 10.0 * dv * dv;
      out[0] = (float)loss;
    }
  }
}

extern "C" void kernel_launch(void* const* d_in, const int* in_sizes, int n_in,
                              void* d_out, int out_size, void* d_ws, size_t ws_size,
                              hipStream_t stream) {
  // setup_inputs order: lat, lon, v, station_lat, station_lon, times
  const float* lat    = (const float*)d_in[0];
  const float* lon    = (const float*)d_in[1];
  const float* v      = (const float*)d_in[2];
  const float* st_lat = (const float*)d_in[3];
  const float* st_lon = (const float*)d_in[4];
  const float* times  = (const float*)d_in[5];
  const int N = in_sizes[3];

  // Tiny O(N) problem (96 KB of input): single workgroup, 32 waves,
  // deterministic, graph-capture safe.
  FIND_LOCATION_43980465111763_kernel<<<1, 1024, 0, stream>>>(
      lat, lon, v, st_lat, st_lon, times, (float*)d_out, N);
}